// ContrastMaximizationLossSpike_74345883894634
// MI455X (gfx1250) — compile-verified
//
#include <hip/hip_runtime.h>

typedef __attribute__((ext_vector_type(2))) float v2f;
typedef __attribute__((ext_vector_type(8))) float v8f;

#define BATCH 4
#define CHANS 64
#define HH    256
#define WW    256
#define NTOT  (CHANS * HH * WW)        // 4,194,304 accumulator elements
#define RED_BLOCKS 2048                // each reduce block handles 2048 elems

// Guaranteed-native device-scope f32 atomic add (no CAS loop).
__device__ __forceinline__ void atomic_add_f32_dev(float* p, float v) {
    asm volatile("global_atomic_add_f32 %0, %1, off scope:SCOPE_DEV"
                 :: "v"(p), "v"(v)
                 : "memory");
}

// ---------------- Kernel 1: zero the accumulator -------------------------
__global__ void k_zero_acc(float4* __restrict__ acc4) {
    int i = blockIdx.x * blockDim.x + threadIdx.x;   // N/4 = 1,048,576 threads
    acc4[i] = make_float4(0.f, 0.f, 0.f, 0.f);
}

// ---------------- Kernel 2: forward bilinear splat -----------------------
// grid.x = BATCH*CHANS*HH (65536 rows), block.x = WW (256)
__global__ void k_splat(const float* __restrict__ flow,
                        const float* __restrict__ spike,
                        float* __restrict__ acc) {
    const int row = blockIdx.x;
    const int w   = threadIdx.x;
    const int b   = row >> 14;            // / (CHANS*HH)
    const int rem = row & 16383;
    const int c   = rem >> 8;             // / HH
    const int h   = rem & 255;

    const float u   = flow[(((b * 2) + 0) * HH + h) * WW + w];
    const float v   = flow[(((b * 2) + 1) * HH + h) * WW + w];
    const float val = spike[((b * CHANS + c) * HH + h) * WW + w];

    const float s  = ((float)c - 31.5f) * 0.015625f;   // (c - mid)/dt, dt = 64
    const float xn = (float)w + u * s;
    const float yn = (float)h + v * s;

    const float x0 = floorf(xn);
    const float y0 = floorf(yn);
    const float wx = xn - x0;
    const float wy = yn - y0;
    const int xi = (int)x0;
    const int yi = (int)y0;

    float* plane = acc + (size_t)c * (HH * WW);

    const float w00 = (1.f - wx) * (1.f - wy) * val;
    const float w10 = wx * (1.f - wy) * val;
    const float w01 = (1.f - wx) * wy * val;
    const float w11 = wx * wy * val;

    if (xi >= 0 && xi < WW && yi >= 0 && yi < HH)
        atomic_add_f32_dev(plane + yi * WW + xi, w00);
    if (xi + 1 >= 0 && xi + 1 < WW && yi >= 0 && yi < HH)
        atomic_add_f32_dev(plane + yi * WW + (xi + 1), w10);
    if (xi >= 0 && xi < WW && yi + 1 >= 0 && yi + 1 < HH)
        atomic_add_f32_dev(plane + (yi + 1) * WW + xi, w01);
    if (xi + 1 >= 0 && xi + 1 < WW && yi + 1 >= 0 && yi + 1 < HH)
        atomic_add_f32_dev(plane + (yi + 1) * WW + (xi + 1), w11);
}

// ---------------- Kernel 3: WMMA-based sum / sumsq reduction -------------
// V_WMMA_F32_16X16X4_F32 with B = ones: D[m,n] = sum_k A[m,k] + C[m,n].
// Packing of the 64 values into A is irrelevant since everything is summed.
// Each wave reduces 64 f32 per WMMA; the C operand accumulates across the loop.
// grid.x = RED_BLOCKS (2048), block.x = 256 (8 waves); block chunk = 2048 elems.
__global__ void k_reduce(const float* __restrict__ acc,
                         float* __restrict__ psum,
                         float* __restrict__ psq) {
    const int lane = threadIdx.x & 31;
    const int wave = threadIdx.x >> 5;

    const size_t base = (size_t)blockIdx.x * 2048;

    v8f cs = {};   // running sum tile
    v8f cq = {};   // running sum-of-squares tile
    v2f ones; ones[0] = 1.f; ones[1] = 1.f;

    #pragma unroll
    for (int t = 0; t < 4; ++t) {
        const size_t idx = base + (size_t)t * 512 + (size_t)wave * 64 + (size_t)lane * 2;
        v2f a = *(const v2f*)(acc + idx);            // 8B-aligned
        v2f a2; a2[0] = a[0] * a[0]; a2[1] = a[1] * a[1];
        cs = __builtin_amdgcn_wmma_f32_16x16x4_f32(false, a,  false, ones,
                                                   (short)0, cs, false, false);
        cq = __builtin_amdgcn_wmma_f32_16x16x4_f32(false, a2, false, ones,
                                                   (short)0, cq, false, false);
    }

    // Per-lane fold of the 8 C VGPRs (rows M=0..7 for lanes 0-15, M=8..15 for 16-31)
    float ls = cs[0] + cs[1] + cs[2] + cs[3] + cs[4] + cs[5] + cs[6] + cs[7];
    float lq = cq[0] + cq[1] + cq[2] + cq[3] + cq[4] + cq[5] + cq[6] + cq[7];
    // Combine the two lane halves (columns are identical, so lane 0 holds the total)
    ls += __shfl_xor(ls, 16, 32);
    lq += __shfl_xor(lq, 16, 32);

    __shared__ float ss[8];
    __shared__ float sq[8];
    if (lane == 0) { ss[wave] = ls; sq[wave] = lq; }
    __syncthreads();
    if (threadIdx.x == 0) {
        float ts = 0.f, tq = 0.f;
        #pragma unroll
        for (int i = 0; i < 8; ++i) { ts += ss[i]; tq += sq[i]; }
        psum[blockIdx.x] = ts;
        psq[blockIdx.x]  = tq;
    }
}

// ---------------- Kernel 4: final f64 combine + loss ---------------------
__global__ void k_final(const float* __restrict__ psum,
                        const float* __restrict__ psq,
                        float* __restrict__ out) {
    __shared__ double ds[256];
    __shared__ double dq[256];
    double s = 0.0, q = 0.0;
    for (int i = threadIdx.x; i < RED_BLOCKS; i += 256) {
        s += (double)psum[i];
        q += (double)psq[i];
    }
    ds[threadIdx.x] = s;
    dq[threadIdx.x] = q;
    __syncthreads();
    for (int stride = 128; stride > 0; stride >>= 1) {
        if (threadIdx.x < stride) {
            ds[threadIdx.x] += ds[threadIdx.x + stride];
            dq[threadIdx.x] += dq[threadIdx.x + stride];
        }
        __syncthreads();
    }
    if (threadIdx.x == 0) {
        const double N = (double)NTOT;
        const double var = (dq[0] - ds[0] * ds[0] / N) / (N - 1.0);
        out[0] = (float)(-var);
    }
}

extern "C" void kernel_launch(void* const* d_in, const int* in_sizes, int n_in,
                              void* d_out, int out_size, void* d_ws, size_t ws_size,
                              hipStream_t stream) {
    const float* flow  = (const float*)d_in[0];   // [4,2,256,256]
    const float* spike = (const float*)d_in[1];   // [4,64,256,256]
    float* out = (float*)d_out;                   // scalar loss

    // Workspace layout: [ accumulator NTOT f32 | psum RED_BLOCKS f32 | psq RED_BLOCKS f32 ]
    float* acc  = (float*)d_ws;
    float* psum = acc + NTOT;
    float* psq  = psum + RED_BLOCKS;

    // 1) zero accumulator (float4 stores): 1,048,576 threads
    k_zero_acc<<<dim3(NTOT / 4 / 256), dim3(256), 0, stream>>>((float4*)acc);

    // 2) splat: one block per (b,c,h) row, one thread per column
    k_splat<<<dim3(BATCH * CHANS * HH), dim3(WW), 0, stream>>>(flow, spike, acc);

    // 3) WMMA reduce to 2048 (sum, sumsq) partials
    k_reduce<<<dim3(RED_BLOCKS), dim3(256), 0, stream>>>(acc, psum, psq);

    // 4) f64 finalize
    k_final<<<dim3(1), dim3(256), 0, stream>>>(psum, psq, out);
}